// SpiralFC_17995912970888
// MI455X (gfx1250) — compile-verified
//
#include <hip/hip_runtime.h>
#include <hip/hip_bf16.h>

typedef __attribute__((ext_vector_type(2))) float v2f;
typedef __attribute__((ext_vector_type(8))) float v8f;
typedef __attribute__((ext_vector_type(4))) unsigned int u32x4;
typedef __attribute__((ext_vector_type(8))) int i32x8;
typedef __attribute__((ext_vector_type(4))) int i32x4;

#define Hd   56
#define Wd   56
#define Cd   512
#define Od   512
#define NPIX (Hd * Wd)

#define TILE_M 128            // output channels per workgroup (8 waves x 16)
#define TILE_N 64             // pixels per workgroup
#define TILE_K 32             // K chunk
#define LROW   (TILE_K + 2)   // 34-float padded rows (TDM pads for us on lwA)

// out[b,o,h,w] = bias[o] + sum_c weight[o,c] * bilinear_shift(x)[b,c,h,w]
// x is NHWC (B,56,56,512); weight is (O=512, C=512); offset is (C,2) = (dy,dx).
__global__ __launch_bounds__(256, 2)
void spiralfc_wmma_f32(const float* __restrict__ x,
                       const float* __restrict__ weight,
                       const float* __restrict__ bias,
                       const float* __restrict__ offset,
                       float* __restrict__ out)
{
    __shared__ float lwA[TILE_M][LROW];   // weight tile, [m][k], filled by TDM
    __shared__ float lwB[TILE_N][LROW];   // shifted-x tile, [n(pixel)][k]

    const int tid  = threadIdx.x;
    const int lane = tid & 31;
    const int wave = tid >> 5;            // 0..7 -> M sub-block of 16

    const int m_base = blockIdx.x * TILE_M;   // 4 M-tiles
    const int p_base = blockIdx.y * TILE_N;   // 49 pixel tiles
    const int b      = blockIdx.z;            // batch

    const float* xb = x + (size_t)b * NPIX * Cd;

    const int lm    = lane & 15;   // 0..15
    const int khalf = lane >> 4;   // 0: K{0,1}  1: K{2,3}

    // ---- accumulators (4 N-subtiles of 16), init with bias.
    //      C/D layout: VGPR r -> M = r + 8*khalf, N = lane%16 ----
    const int o_w = m_base + wave * 16;
    const float bv_init[1] = {0};
    v8f acc0, acc1, acc2, acc3;
#pragma unroll
    for (int r = 0; r < 8; ++r) {
        float bv = bias[o_w + r + 8 * khalf];
        acc0[r] = bv; acc1[r] = bv; acc2[r] = bv; acc3[r] = bv;
    }
    (void)bv_init;

    // ---- constant part of the TDM descriptor (weight tile DMA) ----
    // Group1: data_size=4B, pad_enable, pad_interval=32 dwords (code 4),
    //         pad_amount=2 dwords (code 1) -> LDS rows of 34 floats.
    //         tensor_dim0=512, tensor_dim1=512, tile=32x128, row stride 512.
    i32x8 g1;
    g1[0] = (int)((2u << 16) | (1u << 20) | (4u << 22) | (1u << 25));
    g1[1] = (int)(512u << 16);   // tensor_dim0[15:0] in [63:48]
    g1[2] = (int)(512u << 16);   // tensor_dim1[15:0] in [95:80]
    g1[3] = (int)(32u  << 16);   // tile_dim0=32 in [127:112]
    g1[4] = (int)TILE_M;         // tile_dim1=128 in [143:128]
    g1[5] = (int)Cd;             // tensor_dim0_stride[31:0]=512
    g1[6] = 0;
    g1[7] = 0;
    const i32x4 gz = {0, 0, 0, 0};          // groups 2/3 unused (2D tensor)
    const unsigned lds_a = (unsigned)(uintptr_t)(void*)&lwA[0][0];

    // Gather-fill assignment for lwB: thread -> (pixel n, 8 consecutive k)
    const int gn = tid >> 2;            // 0..63 pixel within tile
    const int gk = (tid & 3) << 3;      // k base (0,8,16,24)
    const int gp = p_base + gn;
    const int gh = gp / Wd;
    const int gw = gp - gh * Wd;

    for (int k0 = 0; k0 < Cd; k0 += TILE_K) {
        // ---- kick off weight-tile DMA (Tensor Data Mover), wave 0 only;
        //      overlaps with the bilinear gather below ----
        if (wave == 0) {
            const unsigned long long ga =
                (unsigned long long)(uintptr_t)(weight + (size_t)m_base * Cd + k0);
            u32x4 g0;
            g0[0] = 1u;                               // count=1, user descriptor
            g0[1] = lds_a;                            // LDS destination
            g0[2] = (unsigned)(ga & 0xffffffffu);     // global_addr[31:0]
            g0[3] = (unsigned)((ga >> 32) & 0x01ffffffu) | (2u << 30); // type=2
#if defined(__clang_major__) && (__clang_major__ >= 23)
            const i32x8 gz8 = {0, 0, 0, 0, 0, 0, 0, 0};
            __builtin_amdgcn_tensor_load_to_lds(g0, g1, gz, gz, gz8, 0);
#else
            __builtin_amdgcn_tensor_load_to_lds(g0, g1, gz, gz, 0);
#endif
        }

        // ---- stage shifted-x tile: general bilinear gather (taps with zero
        //      fractional weight skipped -> 1 load/elem for integer offsets) ----
#pragma unroll
        for (int e = 0; e < 8; ++e) {
            const int kk = gk + e;
            const int c  = k0 + kk;
            const float dy = offset[2 * c + 0];
            const float dx = offset[2 * c + 1];
            const float py = (float)gh + dy;
            const float px = (float)gw + dx;
            const float fy = floorf(py), fx = floorf(px);
            const int   iy0 = (int)fy,   ix0 = (int)fx;
            const float wy1 = py - fy,   wx1 = px - fx;
            const float wy0 = 1.0f - wy1, wx0 = 1.0f - wx1;

            auto tap = [&](int iy, int ix) -> float {
                if ((unsigned)iy < (unsigned)Hd && (unsigned)ix < (unsigned)Wd)
                    return xb[((size_t)iy * Wd + ix) * Cd + c];
                return 0.0f;
            };

            float v = tap(iy0, ix0) * (wy0 * wx0);
            if (wx1 != 0.0f) v += tap(iy0, ix0 + 1) * (wy0 * wx1);
            if (wy1 != 0.0f) {
                v += tap(iy0 + 1, ix0) * (wy1 * wx0);
                if (wx1 != 0.0f) v += tap(iy0 + 1, ix0 + 1) * (wy1 * wx1);
            }
            lwB[gn][kk] = v;
        }

        // DMA + gather must both land before compute
        __builtin_amdgcn_s_wait_tensorcnt(0);
        __syncthreads();

        // ---- WMMA: D = A(16x4) x B(4x16) + C, fp32 in / fp32 out.
        // A frag: lane<16 -> {A[M][k],A[M][k+1]}, lane>=16 -> {A[M][k+2],A[M][k+3]}
        // B frag: lane<16 -> {B[k][N],B[k+1][N]}, lane>=16 -> {B[k+2][N],B[k+3][N]}
        const int a_row = wave * 16 + lm;
#pragma unroll
        for (int k4 = 0; k4 < TILE_K; k4 += 4) {
            const int kc = k4 + 2 * khalf;
            v2f a  = *(const v2f*)&lwA[a_row][kc];
            v2f b0 = *(const v2f*)&lwB[lm][kc];
            v2f b1 = *(const v2f*)&lwB[lm + 16][kc];
            v2f b2 = *(const v2f*)&lwB[lm + 32][kc];
            v2f b3 = *(const v2f*)&lwB[lm + 48][kc];
            acc0 = __builtin_amdgcn_wmma_f32_16x16x4_f32(
                       false, a, false, b0, (short)0, acc0, false, false);
            acc1 = __builtin_amdgcn_wmma_f32_16x16x4_f32(
                       false, a, false, b1, (short)0, acc1, false, false);
            acc2 = __builtin_amdgcn_wmma_f32_16x16x4_f32(
                       false, a, false, b2, (short)0, acc2, false, false);
            acc3 = __builtin_amdgcn_wmma_f32_16x16x4_f32(
                       false, a, false, b3, (short)0, acc3, false, false);
        }

        __syncthreads();
    }

    // ---- store: out is (B, O, 56, 56); 16 consecutive pixels per lane group
    //      -> 64B coalesced segments ----
    {
        const int p0 = p_base + lm;
#pragma unroll
        for (int r = 0; r < 8; ++r) {
            const int o = o_w + r + 8 * khalf;
            const size_t obase = ((size_t)b * Od + o) * NPIX + p0;
            out[obase]      = acc0[r];
            out[obase + 16] = acc1[r];
            out[obase + 32] = acc2[r];
            out[obase + 48] = acc3[r];
        }
    }
}

extern "C" void kernel_launch(void* const* d_in, const int* in_sizes, int n_in,
                              void* d_out, int out_size, void* d_ws, size_t ws_size,
                              hipStream_t stream) {
    const float* x      = (const float*)d_in[0];
    const float* weight = (const float*)d_in[1];
    const float* bias   = (const float*)d_in[2];
    const float* offset = (const float*)d_in[3];
    float* out = (float*)d_out;

    const int batch = in_sizes[0] / (NPIX * Cd);   // = 32

    dim3 grid(Od / TILE_M, NPIX / TILE_N, batch);  // (4, 49, 32)
    spiralfc_wmma_f32<<<grid, 256, 0, stream>>>(x, weight, bias, offset, out);
}